// Hopf_56246891708596
// MI455X (gfx1250) — compile-verified
//
#include <hip/hip_runtime.h>

// Hopf oscillator scan for MI455X (gfx1250, wave32).
//   B=32, T=1000, D=512 -> 16384 independent sequential chains.
// Memory-bound (~262 MB @ 23.3 TB/s ~= 11 us floor). No matmul -> no WMMA.
// CDNA5 path used: double-buffered GLOBAL_LOAD_ASYNC_TO_LDS_B32 + s_wait_asynccnt.

#define BATCH   32
#define TSTEPS  1000
#define DIM     512
#define BLOCK   128                  // 4 wave32 waves / workgroup
#define DCHUNKS (DIM / BLOCK)        // 4 d-chunks per batch row -> 128 blocks
#define TCHUNK  8                    // timesteps staged per async chunk
#define NCHUNK  (TSTEPS / TCHUNK)    // 125
#define NTOT    (BATCH * TSTEPS * DIM)

// Low 32 bits of a generic pointer to LDS == wave-relative DS byte offset
// (ISA 10.2: low 32 bits are the offset, relative to wave's LDS allocation).
__device__ __forceinline__ unsigned lds_off32(const void* p) {
  return (unsigned)(unsigned long long)p;
}

// GLOBAL_LOAD_ASYNC_TO_LDS_B32: per-lane 4B copy, tracked by ASYNCcnt.
// vdst VGPR carries the LDS byte offset, vaddr the 64-bit global address.
__device__ __forceinline__ void async_copy_f32(unsigned lds_byte_off, const float* gsrc) {
  asm volatile("global_load_async_to_lds_b32 %0, %1, off"
               :
               : "v"(lds_byte_off), "v"(gsrc)
               : "memory");
}

template <int N>
__device__ __forceinline__ void wait_asynccnt_le() {
  asm volatile("s_wait_asynccnt %0" : : "n"(N) : "memory");
}

__global__ __launch_bounds__(BLOCK) void hopf_scan_kernel(
    const float* __restrict__ Xr, const float* __restrict__ Xi,
    const float* __restrict__ omegas, float* __restrict__ out) {
  // Double-buffered staging tile: [buf][timestep][xr(0..127) | xi(128..255)]
  __shared__ float tile[2][TCHUNK][2 * BLOCK];

  const int tid  = threadIdx.x;
  const int b    = blockIdx.x / DCHUNKS;
  const int d    = (blockIdx.x % DCHUNKS) * BLOCK + tid;
  const int base = b * (TSTEPS * DIM) + d;

  const float* pR = Xr + base;
  const float* pI = Xi + base;
  float* oR = out + base;           // z_real
  float* oI = out + NTOT + base;    // z_imag (tuple concat order)

  // omega = (w * (MAX-MIN) + MIN) * 2*pi ; MAX-MIN = 10.0
  const float omega = (omegas[d] * 10.0f + 0.1f) * 6.2831853071795864769f;

  float r = 1.0f, phi = 0.0f;
  float c = 1.0f, s = 0.0f;         // cos(phi0)=1, sin(phi0)=0

  // Prefetch chunk 0 into buffer 0 (16 async ops per wave, ASYNCcnt <= 16).
#pragma unroll
  for (int tt = 0; tt < TCHUNK; ++tt) {
    async_copy_f32(lds_off32(&tile[0][tt][tid]),         pR + tt * DIM);
    async_copy_f32(lds_off32(&tile[0][tt][BLOCK + tid]), pI + tt * DIM);
  }

  for (int chunk = 0; chunk < NCHUNK; ++chunk) {
    const int buf = chunk & 1;

    if (chunk + 1 < NCHUNK) {
      // Issue next chunk's async loads, then wait until only those remain
      // in flight (current chunk's 16 ops complete -> its LDS data valid).
      const int t1   = (chunk + 1) * TCHUNK;
      const int nbuf = buf ^ 1;
#pragma unroll
      for (int tt = 0; tt < TCHUNK; ++tt) {
        async_copy_f32(lds_off32(&tile[nbuf][tt][tid]),         pR + (t1 + tt) * DIM);
        async_copy_f32(lds_off32(&tile[nbuf][tt][BLOCK + tid]), pI + (t1 + tt) * DIM);
      }
      wait_asynccnt_le<2 * TCHUNK>();
    } else {
      wait_asynccnt_le<0>();
    }
    // Each lane reads only LDS slots its own wave's async ops wrote,
    // so no workgroup barrier is needed.

    const int t0 = chunk * TCHUNK;
#pragma unroll
    for (int tt = 0; tt < TCHUNK; ++tt) {
      const float xr = tile[buf][tt][tid];
      const float xi = tile[buf][tt][BLOCK + tid];

      // input terms use cos/sin of the *old* phi (== c,s carried over)
      const float input_r   = 5.0f * xr * c;
      const float input_phi = 5.0f * xi * s;

      r   = r + ((1.0f - r * r) * r + input_r) * 0.001f;   // dt = 1e-3
      phi = phi + (omega - input_phi) * 0.001f;

      // sincos of the *new* phi: emitted with z, reused as next step's input
      c = __cosf(phi);
      s = __sinf(phi);

      const int idx = (t0 + tt) * DIM;
      // 262 MB of streamed input+output > 192 MB L2 -> non-temporal stores
      __builtin_nontemporal_store(r * c, oR + idx);
      __builtin_nontemporal_store(r * s, oI + idx);
    }
  }
}

extern "C" void kernel_launch(void* const* d_in, const int* in_sizes, int n_in,
                              void* d_out, int out_size, void* d_ws, size_t ws_size,
                              hipStream_t stream) {
  (void)in_sizes; (void)n_in; (void)out_size; (void)d_ws; (void)ws_size;
  const float* Xr = (const float*)d_in[0];
  const float* Xi = (const float*)d_in[1];
  const float* om = (const float*)d_in[2];
  float* out = (float*)d_out;
  hopf_scan_kernel<<<dim3(BATCH * DCHUNKS), dim3(BLOCK), 0, stream>>>(Xr, Xi, om, out);
}